// Experiment16_55722905698572
// MI455X (gfx1250) — compile-verified
//
#include <hip/hip_runtime.h>
#include <cstddef>
#include <math.h>

#define B_ 128
#define S_ 256
#define V_ 50000
#define E_ 300
#define H_ 512
#define C_ 5
#define F_ (2*H_ + E_)      // 1324
#define KP_ ((H_ + E_)/4)   // 203 WMMA K-steps (K=4 each) per output tile per step

typedef __attribute__((ext_vector_type(2))) float v2f;
typedef __attribute__((ext_vector_type(8))) float v8f;

#define WMMA_F32(a, b, c) \
    __builtin_amdgcn_wmma_f32_16x16x4_f32(false, (a), false, (b), (short)0, (c), false, false)

// ---------------------------------------------------------------------------
// Pack W [M][K] row-major into K-pair-interleaved transposed form:
//   P[j][m][2] = { W[m][2j], W[m][2j+1] }   (j = pair index over K)
// A WMMA A-fragment becomes one coalesced b64 load per lane. The harness
// workspace places pack(W) and pack(Ws) contiguously, forming a single
// [ (H+E)/2 ][M][2] combined matrix for the unified K loop.
// ---------------------------------------------------------------------------
__global__ void pack_weights(const float* __restrict__ W, float* __restrict__ P,
                             int M, int K) {
    int t = blockIdx.x * blockDim.x + threadIdx.x;       // over (K/2)*M
    int total = (K >> 1) * M;
    if (t >= total) return;
    int j = t / M;
    int m = t - j * M;
    P[2*t + 0] = W[(size_t)m*K + 2*j + 0];
    P[2*t + 1] = W[(size_t)m*K + 2*j + 1];
}

// ---------------------------------------------------------------------------
// Embedding gather into [S][E][B] layout.
// ---------------------------------------------------------------------------
__global__ void gather_embed(const int* __restrict__ x, const float* __restrict__ tab,
                             float* __restrict__ embS) {
    int t = blockIdx.x * blockDim.x + threadIdx.x;       // t = s*E*B + e*B + b
    if (t >= S_*E_*B_) return;
    int b  = t & (B_ - 1);
    int se = t >> 7;
    int e  = se % E_;
    int s  = se / E_;
    int row = x[(size_t)b*S_ + s];
    embS[t] = tab[(size_t)row*E_ + e];
}

// ---------------------------------------------------------------------------
// Sequential recurrence. One workgroup = (direction, 16-batch-column group),
// 32 waves, each owning one 16-row M-tile. Per step one unified K loop:
//   acc(16x16) += [W|Ws](16 x 812) x [c;e](812 x 16)   (203 f32 WMMAs)
// [c;e] lives in a single pair-packed LDS array (B-frag = one ds_load_b64,
// conflict-free). The K loop is software-pipelined with distance 2 so WMMAs
// overlap outstanding global/LDS loads instead of s_wait 0 on every op.
// Next step's e-panel is prefetched into registers during the WMMA loop.
// ---------------------------------------------------------------------------
__global__ __launch_bounds__(1024)
void recur_kernel(const float* __restrict__ PW_l,   // combined pack [406][512][2]
                  const float* __restrict__ PW_r,
                  const float* __restrict__ cl, const float* __restrict__ cr,
                  const float* __restrict__ embS,
                  float* __restrict__ leftbuf, float* __restrict__ rightbuf) {
    const int bg  = blockIdx.x;
    const int dir = blockIdx.y;

    const v2f*  __restrict__ Wp  = (const v2f*)(dir ? PW_r : PW_l);
    const float* __restrict__ cv = dir ? cr : cl;
    float* __restrict__ outb     = dir ? rightbuf : leftbuf;

    // pair-packed [kp][n][2]: rows 0..H/2-1 = c-state, rows H/4..H/4+E/2-1 = e-panel (float offset H*8)
    __shared__ float kc_lds[(H_ + E_) * 16];        // 50.75 KB

    const int tid   = threadIdx.x;
    const int mtile = tid >> 5;
    const int lane  = tid & 31;
    const int ncol  = lane & 15;
    const int ksel  = lane >> 4;                    // pair-select within K-quad
    const int mrow  = mtile * 16 + ncol;

    const v2f* __restrict__ ap = Wp + (size_t)ksel*H_ + mrow;       // ap[i*1024]
    const v2f* __restrict__ bp = ((const v2f*)kc_lds) + ksel*16 + ncol; // bp[i*32]

    // ---- init state c = broadcast(cv); emit initial-context slot
    const int slot0 = dir ? (S_ - 1) : 0;
    for (int i = tid; i < H_*16; i += 1024) {
        int h = i >> 4, n = i & 15;
        float v = cv[h];
        kc_lds[((h >> 1) << 5) + (n << 1) + (h & 1)] = v;
        outb[(size_t)h*S_*B_ + (size_t)slot0*B_ + bg*16 + n] = v;
    }
    // ---- stage e-panel for t = 0
    {
        int s0 = dir ? (S_ - 3) : 0;
        const float* esrc = embS + (size_t)s0*E_*B_ + bg*16;
        for (int i = tid; i < E_*16; i += 1024) {
            int e = i >> 4, n = i & 15;
            kc_lds[(H_*8) + ((e >> 1) << 5) + (n << 1) + (e & 1)] = esrc[(size_t)e*B_ + n];
        }
    }
    __syncthreads();

    for (int t = 0; t < S_ - 1; ++t) {
        const int slot = dir ? t : (t + 1);

        // -- prefetch NEXT step's e-panel into registers (hidden by WMMAs)
        float er[5];
        const int t1 = t + 1;
        if (t1 < S_ - 1) {
            int s_nx = dir ? ((t1 == S_-2) ? (S_-1) : (S_-3 - t1)) : t1;
            const float* esrc = embS + (size_t)s_nx*E_*B_ + bg*16;
            #pragma unroll
            for (int j = 0; j < 5; ++j) {
                int i = tid + j*1024;
                if (i < E_*16) er[j] = esrc[(size_t)(i >> 4)*B_ + (i & 15)];
            }
        }

        // -- unified K loop, software-pipelined (distance 2)
        v8f acc = {};
        v2f a0 = ap[0],    b0 = bp[0];
        v2f a1 = ap[1024], b1 = bp[32];
        int i = 0;
        for (; i + 4 <= KP_; i += 2) {
            v2f an0 = ap[(i+2)*1024], bn0 = bp[(i+2)*32];
            v2f an1 = ap[(i+3)*1024], bn1 = bp[(i+3)*32];
            acc = WMMA_F32(a0, b0, acc);
            acc = WMMA_F32(a1, b1, acc);
            a0 = an0; b0 = bn0; a1 = an1; b1 = bn1;
        }
        acc = WMMA_F32(a0, b0, acc);            // index KP_-3
        acc = WMMA_F32(a1, b1, acc);            // index KP_-2
        {
            v2f a2 = ap[(KP_-1)*1024], b2 = bp[(KP_-1)*32];
            acc = WMMA_F32(a2, b2, acc);        // index KP_-1 (203 is odd)
        }

        #pragma unroll
        for (int v = 0; v < 8; ++v) acc[v] = fmaxf(acc[v], 0.0f);   // relu

        __syncthreads();    // all waves done reading c/e before overwrite

        // C/D layout: VGPR v -> M = v + 8*ksel within the tile
        const int hbase = mtile*16 + (ksel << 3);
        #pragma unroll
        for (int v = 0; v < 8; ++v) {
            int h = hbase + v;
            kc_lds[((h >> 1) << 5) + (ncol << 1) + (h & 1)] = acc[v];
            outb[(size_t)h*S_*B_ + (size_t)slot*B_ + bg*16 + ncol] = acc[v];
        }
        // commit prefetched e-panel for step t+1
        if (t1 < S_ - 1) {
            #pragma unroll
            for (int j = 0; j < 5; ++j) {
                int i2 = tid + j*1024;
                if (i2 < E_*16) {
                    int e = i2 >> 4, n = i2 & 15;
                    kc_lds[(H_*8) + ((e >> 1) << 5) + (n << 1) + (e & 1)] = er[j];
                }
            }
        }
        __syncthreads();
    }
}

// ---------------------------------------------------------------------------
// Max-over-S kernels, replicating the reference's reshape/view semantics:
//   left_repr[b,s,h]  = leftflat [b*S*H + s*H + h]
//   right_repr[b,s,h] = rightflat[h*S*B + b*S + s]
// ---------------------------------------------------------------------------
__global__ void max_left(const float* __restrict__ lf, float* __restrict__ mx) {
    int t = blockIdx.x*blockDim.x + threadIdx.x;   // b*H + h (h fastest)
    if (t >= B_*H_) return;
    int h = t & (H_ - 1), b = t >> 9;
    size_t base = (size_t)b*S_*H_ + h;
    float m = lf[base];
    for (int s = 1; s < S_; ++s) m = fmaxf(m, lf[base + (size_t)s*H_]);
    mx[(size_t)b*F_ + h] = m;
}

__global__ void max_emb(const float* __restrict__ embS, float* __restrict__ mx) {
    int t = blockIdx.x*blockDim.x + threadIdx.x;   // e*B + b (b fastest)
    if (t >= B_*E_) return;
    int b = t & (B_ - 1), e = t >> 7;
    size_t base = (size_t)e*B_ + b;
    float m = embS[base];
    for (int s = 1; s < S_; ++s) m = fmaxf(m, embS[base + (size_t)s*E_*B_]);
    mx[(size_t)b*F_ + H_ + e] = m;
}

__global__ void max_right(const float* __restrict__ rf, float* __restrict__ mx) {
    int t = blockIdx.x*blockDim.x + threadIdx.x;   // b*H + h
    if (t >= B_*H_) return;
    int h = t & (H_ - 1), b = t >> 9;
    const float4* p = (const float4*)(rf + (size_t)h*S_*B_ + (size_t)b*S_);
    float m = -INFINITY;
    for (int i = 0; i < S_/4; ++i) {
        float4 v = p[i];
        m = fmaxf(m, fmaxf(fmaxf(v.x, v.y), fmaxf(v.z, v.w)));
    }
    mx[(size_t)b*F_ + H_ + E_ + h] = m;
}

// ---------------------------------------------------------------------------
// Tiny FC: out[b][c] = sum_j mx[b][j]*w[c][j] + bias[c] (deterministic reduce)
// ---------------------------------------------------------------------------
__global__ void fc_kernel(const float* __restrict__ mx, const float* __restrict__ w,
                          const float* __restrict__ bias, float* __restrict__ out) {
    int b = blockIdx.x;
    int tid = threadIdx.x;                       // 128 threads
    float s[C_] = {0.f, 0.f, 0.f, 0.f, 0.f};
    for (int j = tid; j < F_; j += 128) {
        float v = mx[(size_t)b*F_ + j];
        #pragma unroll
        for (int c = 0; c < C_; ++c) s[c] += v * w[(size_t)c*F_ + j];
    }
    __shared__ float red[C_][128];
    #pragma unroll
    for (int c = 0; c < C_; ++c) red[c][tid] = s[c];
    __syncthreads();
    for (int off = 64; off > 0; off >>= 1) {
        if (tid < off)
            #pragma unroll
            for (int c = 0; c < C_; ++c) red[c][tid] += red[c][tid + off];
        __syncthreads();
    }
    if (tid < C_) out[(size_t)b*C_ + tid] = red[tid][0] + bias[tid];
}

// ---------------------------------------------------------------------------
extern "C" void kernel_launch(void* const* d_in, const int* in_sizes, int n_in,
                              void* d_out, int out_size, void* d_ws, size_t ws_size,
                              hipStream_t stream) {
    (void)in_sizes; (void)n_in; (void)out_size; (void)ws_size;

    const int*   x   = (const int*)  d_in[0];
    const float* tab = (const float*)d_in[1];
    const float* Wl  = (const float*)d_in[2];
    const float* Wsl = (const float*)d_in[3];
    const float* Wr  = (const float*)d_in[4];
    const float* Wsr = (const float*)d_in[5];
    const float* cl  = (const float*)d_in[6];
    const float* cr  = (const float*)d_in[7];
    const float* fcw = (const float*)d_in[8];
    const float* fcb = (const float*)d_in[9];
    float* out = (float*)d_out;

    // workspace carve-up (~170 MiB). NOTE: pWl+pWsl (and pWr+pWsr) are
    // intentionally contiguous -> combined [ (H+E)/2 ][H][2] weight matrix.
    float* ws = (float*)d_ws;
    size_t o = 0;
    float* embS = ws + o; o += (size_t)S_*E_*B_;   // [S][E][B]
    float* lbuf = ws + o; o += (size_t)H_*S_*B_;   // [H][S][B]
    float* rbuf = ws + o; o += (size_t)H_*S_*B_;   // [H][S][B]
    float* pWl  = ws + o; o += (size_t)H_*H_;      // combined base (left)
    float* pWsl = ws + o; o += (size_t)E_*H_;
    float* pWr  = ws + o; o += (size_t)H_*H_;      // combined base (right)
    float* pWsr = ws + o; o += (size_t)E_*H_;
    float* mx   = ws + o; o += (size_t)B_*F_;

    pack_weights<<<((H_/2)*H_ + 255)/256, 256, 0, stream>>>(Wl,  pWl,  H_, H_);
    pack_weights<<<((E_/2)*H_ + 255)/256, 256, 0, stream>>>(Wsl, pWsl, H_, E_);
    pack_weights<<<((H_/2)*H_ + 255)/256, 256, 0, stream>>>(Wr,  pWr,  H_, H_);
    pack_weights<<<((E_/2)*H_ + 255)/256, 256, 0, stream>>>(Wsr, pWsr, H_, E_);

    gather_embed<<<(S_*E_*B_)/256, 256, 0, stream>>>(x, tab, embS);

    recur_kernel<<<dim3(B_/16, 2), 1024, 0, stream>>>(
        pWl, pWr, cl, cr, embS, lbuf, rbuf);

    max_left <<<(B_*H_)/256, 256, 0, stream>>>(lbuf, mx);
    max_emb  <<<(B_*E_)/256, 256, 0, stream>>>(embS, mx);
    max_right<<<(B_*H_)/256, 256, 0, stream>>>(rbuf, mx);

    fc_kernel<<<B_, 128, 0, stream>>>(mx, fcw, fcb, out);
}